// BoundaryLoss_72679436583651
// MI455X (gfx1250) — compile-verified
//
#include <hip/hip_runtime.h>
#include <hip/hip_bf16.h>
#include <math.h>

#define HH 512
#define WW 512
#define BB 8
#define CC 4
#define SEL 1

typedef float v2f __attribute__((ext_vector_type(2)));
typedef float v8f __attribute__((ext_vector_type(8)));

// ---------------- init small accumulators ----------------
__global__ void bl_init(unsigned* __restrict__ minU, unsigned* __restrict__ maxU,
                        int* __restrict__ flags) {
    int t = threadIdx.x;
    if (t < 16) { minU[t] = 0x7F800000u; maxU[t] = 0u; }  // +inf / 0 (all d2 >= 0)
    if (t < 8)  flags[t] = 0;
}

// ---------------- column EDT pass (vertical min-propagation) ----------------
// one thread per (batch, transform, column); writes g^2 into gbuf[bt][i][col]
__global__ void bl_edt_cols(const float* __restrict__ y_true, float* __restrict__ gbuf,
                            int* __restrict__ flags) {
    int idx = blockIdx.x * blockDim.x + threadIdx.x;     // 0..8191
    int col = idx & (WW - 1);
    int bt  = idx >> 9;                                   // b*2 + t
    int t   = bt & 1;
    int b   = bt >> 1;
    const float* yt = y_true + ((size_t)b * CC + SEL) * HH * WW + col;
    float* g = gbuf + (size_t)bt * HH * WW + col;

    float prev = 1.0e9f;
    int any = 0;
    for (int i = 0; i < HH; ++i) {
        if (i + 8 < HH) __builtin_prefetch(&yt[(size_t)(i + 8) * WW], 0, 1);
        float v = yt[(size_t)i * WW];
        int m = (v != 0.0f);
        if (t) m ^= 1;
        if (t == 0) any |= m;
        float init = m ? 1024.0f : 0.0f;                  // BIG = H + W = 1024
        float gi = fminf(init, prev + 1.0f);
        g[(size_t)i * WW] = gi;
        prev = gi;
    }
    prev = 1.0e9f;
    for (int i = HH - 1; i >= 0; --i) {
        float gi = fminf(g[(size_t)i * WW], prev + 1.0f);
        prev = gi;
        g[(size_t)i * WW] = gi * gi;                      // store g^2 (exact integer in fp32)
    }
    if (t == 0 && any) atomicOr(&flags[b], 1);
}

// ---------------- row EDT pass: d2[j] = min_k g2[k] + (j-k)^2 ----------------
// one block per (b,t,row); g2 row staged in LDS; also per-image min/max of d2
__global__ void bl_edt_rows(float* __restrict__ gbuf, unsigned* __restrict__ minU,
                            unsigned* __restrict__ maxU) {
    __shared__ float sg[WW];
    __shared__ float sred[256];
    int row = blockIdx.x & (HH - 1);
    int bt  = blockIdx.x >> 9;
    float* g = gbuf + ((size_t)bt * HH + row) * WW;
    int tid = threadIdx.x;

    // widened LDS fill: one b64 per thread
    {
        float2 v = ((const float2*)g)[tid];
        sg[2 * tid]     = v.x;
        sg[2 * tid + 1] = v.y;
    }
    __syncthreads();

    int   j0 = tid, j1 = tid + 256;
    float best0 = 3.4e38f, best1 = 3.4e38f;
    // float-tracked differences: values <= 512 are exact in fp32
    float d0 = (float)j0;
    float d1 = (float)j1;
#pragma unroll 8
    for (int k = 0; k < WW; ++k) {
        float gk = sg[k];
        best0 = fminf(best0, fmaf(d0, d0, gk));
        best1 = fminf(best1, fmaf(d1, d1, gk));
        d0 -= 1.0f;
        d1 -= 1.0f;
    }
    g[j0] = best0;                                        // overwrite with d2 (LDS copy already taken)
    g[j1] = best1;

    float lmin = fminf(best0, best1);
    float lmax = fmaxf(best0, best1);
    sred[tid] = lmin; __syncthreads();
    for (int s = 128; s > 0; s >>= 1) {
        if (tid < s) sred[tid] = fminf(sred[tid], sred[tid + s]);
        __syncthreads();
    }
    float bmin = sred[0];
    __syncthreads();
    sred[tid] = lmax; __syncthreads();
    for (int s = 128; s > 0; s >>= 1) {
        if (tid < s) sred[tid] = fmaxf(sred[tid], sred[tid + s]);
        __syncthreads();
    }
    if (tid == 0) {
        // d2 >= 0 so IEEE bit pattern is monotone as unsigned
        atomicMin(&minU[bt], __float_as_uint(bmin));
        atomicMax(&maxU[bt], __float_as_uint(sred[0]));
    }
}

// ---------------- fused sdf + sigmoid reduction (WMMA wave-reduce) ----------------
__global__ void bl_fuse_reduce(const float* __restrict__ y_pred, const float* __restrict__ y_true,
                               const float* __restrict__ gbuf,
                               const unsigned* __restrict__ minU, const unsigned* __restrict__ maxU,
                               const int* __restrict__ flags,
                               float* __restrict__ blockSums) {
    const size_t NTH = (size_t)2048 * 256;                // total threads; 4 pixels each
    size_t gt = (size_t)blockIdx.x * blockDim.x + threadIdx.x;
    float acc0 = 0.0f, acc1 = 0.0f;

    for (int it = 0; it < 4; ++it) {
        size_t p = gt + (size_t)it * NTH;                 // pixel index over B*H*W
        int b = (int)(p >> 18);                           // / (512*512)
        int r = (int)((p >> 9) & (HH - 1));
        int c = (int)(p & (WW - 1));

        float d2p = gbuf[((size_t)(2 * b) * HH + r) * WW + c];
        float d2n = gbuf[((size_t)(2 * b + 1) * HH + r) * WW + c];
        float posdis = sqrtf(d2p), negdis = sqrtf(d2n);
        float pmin = sqrtf(__uint_as_float(minU[2 * b]));
        float pmax = sqrtf(__uint_as_float(maxU[2 * b]));
        float nmin = sqrtf(__uint_as_float(minU[2 * b + 1]));
        float nmax = sqrtf(__uint_as_float(maxU[2 * b + 1]));
        float sdf = (negdis - nmin) / (nmax - nmin) - (posdis - pmin) / (pmax - pmin);

        // inner boundary (connectivity-1, pad=True): fg pixel with an in-image bg 4-neighbor.
        // Branchless: clamp neighbor indices (always in-bounds loads, overlapped in the
        // memory pipeline), then predicate with in-image tests via selects.
        const float* yt = y_true + ((size_t)b * CC + SEL) * HH * WW;
        int rm = r > 0 ? r - 1 : 0;
        int rp = r < HH - 1 ? r + 1 : HH - 1;
        int cm = c > 0 ? c - 1 : 0;
        int cp = c < WW - 1 ? c + 1 : WW - 1;
        float vc = yt[(size_t)r * WW + c];
        float vu = yt[(size_t)rm * WW + c];
        float vd = yt[(size_t)rp * WW + c];
        float vl = yt[(size_t)r * WW + cm];
        float vr = yt[(size_t)r * WW + cp];
        bool bg = ((r > 0)      & (vu == 0.0f)) |
                  ((r < HH - 1) & (vd == 0.0f)) |
                  ((c > 0)      & (vl == 0.0f)) |
                  ((c < WW - 1) & (vr == 0.0f));
        bool zero = ((vc != 0.0f) & bg) | (flags[b] == 0);
        sdf = zero ? 0.0f : sdf;

        float s = 0.0f;
#pragma unroll
        for (int ch = 0; ch < CC; ++ch) {
            float x = y_pred[(((size_t)b * CC + ch) * HH + r) * WW + c];
            s += 1.0f / (1.0f + __expf(-x));
        }
        float contrib = sdf * s;
        if (it & 1) acc1 += contrib; else acc0 += contrib;
    }

    // wave32 reduction of 64 fp32 partials via V_WMMA_F32_16X16X4_F32 with B = ones:
    // D[m,n] = sum_k A[m,k]  ->  16 row sums; column n lives in lanes n (M0..7) and n+16 (M8..15)
    v2f a;   a.x = acc0;  a.y = acc1;
    v2f one; one.x = 1.0f; one.y = 1.0f;
    v8f cz = {};
    v8f d = __builtin_amdgcn_wmma_f32_16x16x4_f32(false, a, false, one, (short)0, cz, false, false);
    float wsum = d[0] + d[1] + d[2] + d[3] + d[4] + d[5] + d[6] + d[7];
    wsum += __shfl_xor(wsum, 16, 32);                     // combine M0..7 and M8..15 halves

    __shared__ float swave[8];
    int lane = threadIdx.x & 31;
    int wave = threadIdx.x >> 5;
    if (lane == 0) swave[wave] = wsum;
    __syncthreads();
    if (threadIdx.x == 0) {
        float bsum = 0.0f;
#pragma unroll
        for (int i = 0; i < 8; ++i) bsum += swave[i];
        blockSums[blockIdx.x] = bsum;
    }
}

// ---------------- deterministic final sum ----------------
__global__ void bl_final(const float* __restrict__ blockSums, float* __restrict__ out) {
    int tid = threadIdx.x;                                // 32 threads
    float s = 0.0f;
    for (int i = 0; i < 64; ++i) s += blockSums[tid * 64 + i];
    for (int off = 16; off > 0; off >>= 1) s += __shfl_xor(s, off, 32);
    if (tid == 0) out[0] = s * (1.0f / 8388608.0f);       // / (B*C*H*W)
}

extern "C" void kernel_launch(void* const* d_in, const int* in_sizes, int n_in,
                              void* d_out, int out_size, void* d_ws, size_t ws_size,
                              hipStream_t stream) {
    const float* y_pred = (const float*)d_in[0];          // (8,4,512,512) f32
    const float* y_true = (const float*)d_in[1];          // (8,4,512,512) f32
    float* out = (float*)d_out;                           // scalar f32

    // workspace layout
    float*    gbuf      = (float*)d_ws;                    // 8*2*512*512 f32 = 16.78 MB
    float*    blockSums = gbuf + (size_t)BB * 2 * HH * WW; // 2048 f32
    unsigned* minU      = (unsigned*)(blockSums + 2048);   // 16
    unsigned* maxU      = minU + 16;                       // 16
    int*      flags     = (int*)(maxU + 16);               // 8

    bl_init<<<1, 32, 0, stream>>>(minU, maxU, flags);
    bl_edt_cols<<<32, 256, 0, stream>>>(y_true, gbuf, flags);
    bl_edt_rows<<<BB * 2 * HH, 256, 0, stream>>>(gbuf, minU, maxU);
    bl_fuse_reduce<<<2048, 256, 0, stream>>>(y_pred, y_true, gbuf, minU, maxU, flags, blockSums);
    bl_final<<<1, 32, 0, stream>>>(blockSums, out);
}